// AttentionDecoder_57131654971751
// MI455X (gfx1250) — compile-verified
//
#include <hip/hip_runtime.h>
#include <math.h>

typedef __attribute__((ext_vector_type(2))) float v2f;
typedef __attribute__((ext_vector_type(4))) float v4f;
typedef __attribute__((ext_vector_type(8))) float v8f;

#define HID   2048
#define VOCAB 50257
#define SEQL  512

// workspace layout (float offsets; all x-vector bases 16B-aligned)
#define WS_XA 0         // cat(emb, h)            [4096]
#define WS_AW 4096      // attn logits -> weights [512]
#define WS_XC 4608      // cat(emb, attn_applied) [4096]
#define WS_XR 8704      // relu(combine)          [2048]
#define WS_GI 10752     // gi                     [6144]
#define WS_GH 16896     // gh                     [6144]
#define WS_HN 23040     // h_new                  [2048]
#define WS_LG 25088     // raw logits             [50257]
#define WS_SC 75346     // logsumexp scalar       [1]

// ---------------------------------------------------------------------------
// prep: gather embedding row, build cat(emb,h) and low half of cat(emb,·),
// zero the attn_applied accumulator half.
__global__ void k_prep(const int* __restrict__ token,
                       const float* __restrict__ hidden,
                       const float* __restrict__ embedding,
                       float* __restrict__ ws) {
  int i = blockIdx.x * blockDim.x + threadIdx.x;
  if (i < HID) {
    float e = embedding[(size_t)token[0] * HID + i];
    ws[WS_XA + i] = e;
    ws[WS_XC + i] = e;
    ws[WS_XA + HID + i] = hidden[i];
    ws[WS_XC + HID + i] = 0.0f;   // accumulated by k_attn_apply
  }
}

// ---------------------------------------------------------------------------
// GEMV y[M] = act(W[M,K] @ x[K] + b[M]) on V_WMMA_F32_16X16X4_F32.
// Per 8-K chunk: one b128 per lane per tile feeds TWO wmma steps.
// K-slot assignment (A layout: lane l -> M=l&15, K pair = 2*(l>>4)):
//   lanes 0-15 load W[m, k0..k0+3], lanes 16-31 load W[m, k0+4..k0+7];
//   step0 uses .xy  -> global k {k0,k0+1,k0+4,k0+5}
//   step1 uses .zw  -> global k {k0+2,k0+3,k0+6,k0+7}
// B mirrors the same assignment from a b128 of x, so every D column equals y.
// C/D layout: lane l, vgpr r -> M = r + 8*(l>>4), N = l&15.
template <int ACT, int TILES>
__device__ __forceinline__
void gemv_core(const float* __restrict__ W, const float* __restrict__ x,
               const float* __restrict__ bias, float* __restrict__ y,
               int M, int K, int block) {
  const int lane  = threadIdx.x & 31;
  const int wave  = threadIdx.x >> 5;
  const int m     = lane & 15;
  const int koff4 = (lane >> 4) * 4;
  const int row0  = block * (TILES * 128) + wave * (TILES * 16);

  const float* wp[TILES];
#pragma unroll
  for (int t = 0; t < TILES; ++t) {
    int r = row0 + t * 16 + m;
    if (r > M - 1) r = M - 1;           // clamp: keeps EXEC all-ones
    wp[t] = W + (size_t)r * K + koff4;
  }
  const float* xp = x + koff4;

  v8f acc[TILES];
#pragma unroll
  for (int t = 0; t < TILES; ++t) acc[t] = (v8f){};

#pragma unroll 2
  for (int k0 = 0; k0 < K; k0 += 8) {
    v4f xv = *(const v4f*)(xp + k0);                                // L2-resident
#pragma unroll
    for (int t = 0; t < TILES; ++t) {
      v4f av = __builtin_nontemporal_load((const v4f*)(wp[t] + k0)); // streamed
      v2f alo = av.xy, ahi = av.zw;
      v2f blo = xv.xy, bhi = xv.zw;
      acc[t] = __builtin_amdgcn_wmma_f32_16x16x4_f32(false, alo, false, blo,
                                                     (short)0, acc[t], false, false);
      acc[t] = __builtin_amdgcn_wmma_f32_16x16x4_f32(false, ahi, false, bhi,
                                                     (short)0, acc[t], false, false);
    }
  }

  // Column N=0 lives in lanes 0 (rows 0..7 of tile) and 16 (rows 8..15).
  if (m == 0) {
    const int half = (lane >> 4) * 8;
#pragma unroll
    for (int t = 0; t < TILES; ++t) {
      int base = row0 + t * 16 + half;
#pragma unroll
      for (int r = 0; r < 8; ++r) {
        int row = base + r;
        if (row < M) {
          float v = acc[t][r] + bias[row];
          if (ACT == 1) v = fmaxf(v, 0.0f);
          y[row] = v;
        }
      }
    }
  }
}

template <int ACT, int TILES>
__global__ __launch_bounds__(256)
void k_gemv_wmma(const float* __restrict__ W, const float* __restrict__ x,
                 const float* __restrict__ bias, float* __restrict__ y,
                 int M, int K) {
  gemv_core<ACT, TILES>(W, x, bias, y, M, K, blockIdx.x);
}

// Fused GRU-gate GEMVs (independent): blocks 0..23 -> gi, 24..47 -> gh.
__global__ __launch_bounds__(256)
void k_gates(const float* __restrict__ w_ih, const float* __restrict__ xr,
             const float* __restrict__ b_ih, float* __restrict__ gi,
             const float* __restrict__ w_hh, const float* __restrict__ h,
             const float* __restrict__ b_hh, float* __restrict__ gh) {
  if (blockIdx.x < 24)
    gemv_core<0, 2>(w_ih, xr, b_ih, gi, 3 * HID, HID, blockIdx.x);
  else
    gemv_core<0, 2>(w_hh, h, b_hh, gh, 3 * HID, HID, blockIdx.x - 24);
}

// ---------------------------------------------------------------------------
// softmax over the 512 attn logits; also emit attn_weights output.
__global__ void k_softmax_attn(float* __restrict__ ws, float* __restrict__ out) {
  __shared__ float red[512];
  int tid = threadIdx.x;
  float v = ws[WS_AW + tid];
  red[tid] = v;
  __syncthreads();
  for (int s = 256; s > 0; s >>= 1) {
    if (tid < s) red[tid] = fmaxf(red[tid], red[tid + s]);
    __syncthreads();
  }
  float mx = red[0];
  __syncthreads();
  float e = expf(v - mx);
  red[tid] = e;
  __syncthreads();
  for (int s = 256; s > 0; s >>= 1) {
    if (tid < s) red[tid] += red[tid + s];
    __syncthreads();
  }
  float w = e / red[0];
  ws[WS_AW + tid] = w;
  out[VOCAB + HID + tid] = w;
}

// ---------------------------------------------------------------------------
// attn_applied[j] = sum_l aw[l]*enc[l,j]; L split across blockIdx.y (32 each),
// coalesced along j, accumulated with float atomics into pre-zeroed ws.
__global__ void k_attn_apply(const float* __restrict__ enc, float* __restrict__ ws) {
  int j  = blockIdx.x * 256 + threadIdx.x;  // 0..2047
  int l0 = blockIdx.y * 32;
  float s = 0.0f;
#pragma unroll 4
  for (int l = 0; l < 32; ++l)
    s += ws[WS_AW + l0 + l] * enc[(size_t)(l0 + l) * HID + j];
  atomicAdd(&ws[WS_XC + HID + j], s);
}

// ---------------------------------------------------------------------------
// GRU cell elementwise (PyTorch r,z,n ordering); emits h_new output too.
__global__ void k_gru(const float* __restrict__ hidden, float* __restrict__ ws,
                      float* __restrict__ out) {
  int i = blockIdx.x * blockDim.x + threadIdx.x;
  if (i >= HID) return;
  float i_r = ws[WS_GI + i], i_z = ws[WS_GI + HID + i], i_n = ws[WS_GI + 2 * HID + i];
  float h_r = ws[WS_GH + i], h_z = ws[WS_GH + HID + i], h_n = ws[WS_GH + 2 * HID + i];
  float r = 1.0f / (1.0f + expf(-(i_r + h_r)));
  float z = 1.0f / (1.0f + expf(-(i_z + h_z)));
  float n = tanhf(i_n + r * h_n);
  float hv = (1.0f - z) * n + z * hidden[i];
  ws[WS_HN + i] = hv;
  out[VOCAB + i] = hv;
}

// ---------------------------------------------------------------------------
// log-sum-exp over the 50257 raw logits (single block, 200KB from L2).
__global__ void k_lse(float* __restrict__ ws) {
  __shared__ float red[1024];
  int tid = threadIdx.x;
  float mx = -3.4e38f;
  for (int i = tid; i < VOCAB; i += 1024) mx = fmaxf(mx, ws[WS_LG + i]);
  red[tid] = mx;
  __syncthreads();
  for (int s = 512; s > 0; s >>= 1) {
    if (tid < s) red[tid] = fmaxf(red[tid], red[tid + s]);
    __syncthreads();
  }
  mx = red[0];
  __syncthreads();
  float sum = 0.0f;
  for (int i = tid; i < VOCAB; i += 1024) sum += expf(ws[WS_LG + i] - mx);
  red[tid] = sum;
  __syncthreads();
  for (int s = 512; s > 0; s >>= 1) {
    if (tid < s) red[tid] += red[tid + s];
    __syncthreads();
  }
  if (tid == 0) ws[WS_SC] = mx + logf(red[0]);
}

__global__ void k_final(const float* __restrict__ ws, float* __restrict__ out) {
  int i = blockIdx.x * blockDim.x + threadIdx.x;
  if (i < VOCAB) out[i] = ws[WS_LG + i] - ws[WS_SC];
}

// ---------------------------------------------------------------------------
extern "C" void kernel_launch(void* const* d_in, const int* in_sizes, int n_in,
                              void* d_out, int out_size, void* d_ws, size_t ws_size,
                              hipStream_t stream) {
  (void)in_sizes; (void)n_in; (void)out_size; (void)ws_size;
  const int*   token  = (const int*)d_in[0];
  const float* hidden = (const float*)d_in[1];
  const float* enc    = (const float*)d_in[2];
  const float* embed  = (const float*)d_in[3];
  const float* attn_W = (const float*)d_in[4];
  const float* attn_b = (const float*)d_in[5];
  const float* comb_W = (const float*)d_in[6];
  const float* comb_b = (const float*)d_in[7];
  const float* w_ih   = (const float*)d_in[8];
  const float* w_hh   = (const float*)d_in[9];
  const float* b_ih   = (const float*)d_in[10];
  const float* b_hh   = (const float*)d_in[11];
  const float* out_W  = (const float*)d_in[12];
  const float* out_b  = (const float*)d_in[13];
  float* out = (float*)d_out;
  float* ws  = (float*)d_ws;

  k_prep<<<8, 256, 0, stream>>>(token, hidden, embed, ws);
  // attn logits: [512 x 4096] @ xa — 256 rows/block
  k_gemv_wmma<0, 2><<<2, 256, 0, stream>>>(attn_W, ws + WS_XA, attn_b, ws + WS_AW, SEQL, 2 * HID);
  k_softmax_attn<<<1, 512, 0, stream>>>(ws, out);
  k_attn_apply<<<dim3(8, 16), 256, 0, stream>>>(enc, ws);
  // combine + relu: [2048 x 4096] @ xc
  k_gemv_wmma<1, 2><<<8, 256, 0, stream>>>(comb_W, ws + WS_XC, comb_b, ws + WS_XR, HID, 2 * HID);
  // GRU gates fused: 2 x [6144 x 2048]
  k_gates<<<48, 256, 0, stream>>>(w_ih, ws + WS_XR, b_ih, ws + WS_GI,
                                  w_hh, hidden, b_hh, ws + WS_GH);
  k_gru<<<8, 256, 0, stream>>>(hidden, ws, out);
  // vocab projection: [50257 x 2048] — the 412 MB bandwidth hog, 197 blocks
  k_gemv_wmma<0, 2><<<(VOCAB + 255) / 256, 256, 0, stream>>>(out_W, ws + WS_HN, out_b, ws + WS_LG, VOCAB, HID);
  k_lse<<<1, 1024, 0, stream>>>(ws);
  k_final<<<(VOCAB + 255) / 256, 256, 0, stream>>>(ws, out);
}